// AveragedHausdorffLoss_36412732735860
// MI455X (gfx1250) — compile-verified
//
#include <hip/hip_runtime.h>
#include <hip/hip_bf16.h>
#include <math.h>

typedef float v2f __attribute__((ext_vector_type(2)));
typedef float v8f __attribute__((ext_vector_type(8)));

#define INF_BITS 0x7F800000u

// ---------------------------------------------------------------------------
// Init kernel: seed global col-min array (uint bit patterns of d^2) with +inf.
// ---------------------------------------------------------------------------
__global__ void ahd_init_kernel(unsigned* __restrict__ colminU, int total) {
    int i = blockIdx.x * blockDim.x + threadIdx.x;
    if (i < total) colminU[i] = INF_BITS;
}

// ---------------------------------------------------------------------------
// Main kernel: one block = 8 waves x 16 rows = 128 rows of set1 for one batch.
// Each wave sweeps all M columns, two 16-wide tiles per iteration; one
// V_WMMA_F32_16X16X4_F32 per 16x16 tile computes the inner products (K=3
// padded to 4).  Minima are tracked on *squared* distances (sqrt is
// monotonic; all sqrts are deferred to the final reduce kernel).  The clamp
// to 0 stays in-loop so the uint-bit-pattern min trick stays order-correct.
// Col-min: every lane issues the LDS uint atomicMin directly (both lane
// halves hit the same address, the DS unit resolves it) — no ds_bpermute,
// no EXEC-masked branch, fully uniform control flow.
// ---------------------------------------------------------------------------
__global__ __launch_bounds__(256) void ahd_main_kernel(
    const float* __restrict__ set1, const float* __restrict__ set2,
    float* __restrict__ rowmin, unsigned* __restrict__ colminU,
    int B, int N, int M) {

    const int tid  = threadIdx.x;
    const int lane = tid & 31;
    const int wave = tid >> 5;              // 0..7
    const int hi   = lane >> 4;             // 0: lanes 0-15, 1: lanes 16-31
    const int l15  = lane & 15;

    const int blocksPerBatch = N / 128;     // 32
    const int b       = blockIdx.x / blocksPerBatch;
    const int chunk   = blockIdx.x % blocksPerBatch;
    const int rowbase = chunk * 128 + wave * 16;

    const float* s1b = set1 + (size_t)b * N * 3;
    const float* s2b = set2 + (size_t)b * M * 3;

    __shared__ unsigned lds_colmin[4096];
    for (int i = tid; i < M; i += 256) lds_colmin[i] = INF_BITS;
    __syncthreads();

    // ---- Load A operand: 16 rows of set1, K padded 3 -> 4 with zeros ----
    // A-matrix 16x4 f32 layout: lanes 0-15 VGPR0=K0 VGPR1=K1;
    //                            lanes 16-31 VGPR0=K2 VGPR1=K3(=0).
    const int myrow = rowbase + l15;
    const float ax = s1b[myrow * 3 + 0];
    const float ay = s1b[myrow * 3 + 1];
    const float az = s1b[myrow * 3 + 2];
    const float x2own = ax * ax + ay * ay + az * az;

    v2f A;
    A.x = hi ? az : ax;
    A.y = hi ? 0.0f : ay;

    // x2 for the 8 output rows this (lane-half, vgpr v) pair sees:
    // C/D row index m' = v + 8*hi.  Lane r (r<16) holds x2 of row r.
    float x2r[8];
#pragma unroll
    for (int v = 0; v < 8; ++v)
        x2r[v] = __shfl(x2own, 8 * hi + v, 32);

    v8f rmin;                                // running min of d^2 per row
#pragma unroll
    for (int v = 0; v < 8; ++v) rmin[v] = __builtin_inff();

    // ---- Sweep all column tiles, 2 tiles (32 cols) per iteration so the
    //      second load's latency hides under the first tile's WMMA+epilogue.
    for (int col = 0; col < M; col += 32) {
        const int n0 = col + l15;
        const int n1 = col + 16 + l15;

        // issue both 96-bit loads up front (compiler emits an s_clause pair)
        const float bx0 = s2b[n0 * 3 + 0];
        const float by0 = s2b[n0 * 3 + 1];
        const float bz0 = s2b[n0 * 3 + 2];
        const float bx1 = s2b[n1 * 3 + 0];
        const float by1 = s2b[n1 * 3 + 1];
        const float bz1 = s2b[n1 * 3 + 2];

        const float y20 = bx0 * bx0 + by0 * by0 + bz0 * bz0;
        const float y21 = bx1 * bx1 + by1 * by1 + bz1 * bz1;

        v2f Bv0, Bv1;                        // B 4x16: per-lane column n,
        Bv0.x = hi ? bz0 : bx0;              // lanes 0-15 K0/K1, 16-31 K2/K3
        Bv0.y = hi ? 0.0f : by0;
        Bv1.x = hi ? bz1 : bx1;
        Bv1.y = hi ? 0.0f : by1;

        v8f c0 = {}, c1 = {};
        c0 = __builtin_amdgcn_wmma_f32_16x16x4_f32(
                false, A, false, Bv0, (short)0, c0, false, false);
        c1 = __builtin_amdgcn_wmma_f32_16x16x4_f32(
                false, A, false, Bv1, (short)0, c1, false, false);

        float cmin0 = __builtin_inff();
        float cmin1 = __builtin_inff();
#pragma unroll
        for (int v = 0; v < 8; ++v) {
            float d20 = fmaxf(x2r[v] + y20 - 2.0f * c0[v], 0.0f);
            float d21 = fmaxf(x2r[v] + y21 - 2.0f * c1[v], 0.0f);
            rmin[v] = fminf(rmin[v], fminf(d20, d21));
            cmin0   = fminf(cmin0, d20);
            cmin1   = fminf(cmin1, d21);
        }
        // All 32 lanes atomic-min into LDS: lanes L and L^16 share a column
        // address; the DS atomic unit performs the cross-half combine for us.
        atomicMin(&lds_colmin[n0], __float_as_uint(cmin0));
        atomicMin(&lds_colmin[n1], __float_as_uint(cmin1));
    }

    // ---- Row-min: reduce across the 16 lanes of each half ----
#pragma unroll
    for (int off = 1; off < 16; off <<= 1) {
#pragma unroll
        for (int v = 0; v < 8; ++v)
            rmin[v] = fminf(rmin[v], __shfl_xor(rmin[v], off, 32));
    }
    if (l15 == 0) {   // lane 0 writes rows m=0..7, lane 16 writes m=8..15
#pragma unroll
        for (int v = 0; v < 8; ++v)
            rowmin[(size_t)b * N + rowbase + 8 * hi + v] = rmin[v]; // d^2
    }

    // ---- Flush block-local col-mins (d^2 bits) to global ----
    __syncthreads();
    for (int i = tid; i < M; i += 256)
        atomicMin(&colminU[(size_t)b * M + i], lds_colmin[i]);
}

// ---------------------------------------------------------------------------
// Final reduction: out = sum(sqrt(rowmin_d2))/(B*N) + sum(sqrt(colmin_d2))/(B*M)
//   (== mean over b of (mean_n rowmin_dist + mean_m colmin_dist))
// Square roots are applied here, off the hot path (only B*(N+M) of them).
// ---------------------------------------------------------------------------
__global__ __launch_bounds__(256) void ahd_reduce_kernel(
    const float* __restrict__ rowmin, const unsigned* __restrict__ colminU,
    float* __restrict__ out, int B, int N, int M) {

    const int tid = threadIdx.x;
    float accR = 0.0f, accC = 0.0f;
    const int totalR = B * N;
    const int totalC = B * M;
    for (int i = tid; i < totalR; i += 256) accR += sqrtf(rowmin[i]);
    for (int i = tid; i < totalC; i += 256) accC += sqrtf(__uint_as_float(colminU[i]));

    float val = accR / (float)totalR + accC / (float)totalC;

    // wave32 tree reduce
#pragma unroll
    for (int off = 16; off > 0; off >>= 1)
        val += __shfl_xor(val, off, 32);

    __shared__ float wsum[8];
    if ((tid & 31) == 0) wsum[tid >> 5] = val;
    __syncthreads();
    if (tid == 0) {
        float s = 0.0f;
#pragma unroll
        for (int w = 0; w < 8; ++w) s += wsum[w];
        out[0] = s;
    }
}

// ---------------------------------------------------------------------------
extern "C" void kernel_launch(void* const* d_in, const int* in_sizes, int n_in,
                              void* d_out, int out_size, void* d_ws, size_t ws_size,
                              hipStream_t stream) {
    const int B = 4;
    const int N = in_sizes[0] / (B * 3);    // 4096
    const int M = in_sizes[1] / (B * 3);    // 4096

    const float* set1 = (const float*)d_in[0];
    const float* set2 = (const float*)d_in[1];
    float* out = (float*)d_out;

    // workspace: rowmin [B*N] floats (d^2) | colmin [B*M] uints (d^2 bits)
    float*    rowmin  = (float*)d_ws;
    unsigned* colminU = (unsigned*)((char*)d_ws + (size_t)B * N * sizeof(float));

    const int totalC = B * M;
    ahd_init_kernel<<<(totalC + 255) / 256, 256, 0, stream>>>(colminU, totalC);

    const int blocks = B * (N / 128);       // 128 blocks, 8 waves each
    ahd_main_kernel<<<blocks, 256, 0, stream>>>(set1, set2, rowmin, colminU, B, N, M);

    ahd_reduce_kernel<<<1, 256, 0, stream>>>(rowmin, colminU, out, B, N, M);
}